// Encoder_35467839930953
// MI455X (gfx1250) — compile-verified
//
#include <hip/hip_runtime.h>

typedef __attribute__((ext_vector_type(2))) float v2f;
typedef __attribute__((ext_vector_type(4))) float v4f;
typedef __attribute__((ext_vector_type(8))) float v8f;

// ---------------- elementwise / graph helpers ----------------

__global__ void fill_kernel(float* __restrict__ p, float v, size_t n) {
  size_t i = (size_t)blockIdx.x * blockDim.x + threadIdx.x;
  if (i < n) p[i] = v;
}

__global__ void relu_kernel(float* __restrict__ p, size_t n) {
  size_t i = (size_t)blockIdx.x * blockDim.x + threadIdx.x;
  if (i < n) p[i] = fmaxf(p[i], 0.0f);
}

// deg[col] += 1 for every edge (deg pre-filled with 1.0 for the self-loop)
__global__ void deg_kernel(const int* __restrict__ ei, float* __restrict__ deg, int E) {
  int e = blockIdx.x * blockDim.x + threadIdx.x;
  if (e < E) atomicAdd(&deg[ei[E + e]], 1.0f);
}

__global__ void rsqrt_kernel(float* __restrict__ p, int n) {
  int i = blockIdx.x * blockDim.x + threadIdx.x;
  if (i < n) p[i] = rsqrtf(p[i]);   // deg >= 1 always (self loop)
}

__global__ void norm_kernel(const int* __restrict__ ei, const float* __restrict__ dinv,
                            float* __restrict__ normE, int E) {
  int e = blockIdx.x * blockDim.x + threadIdx.x;
  if (e < E) normE[e] = dinv[ei[e]] * dinv[ei[E + e]];
}

// agg[i,f] = dinv[i]^2 * m[i,f] + bias[f]   (self-loop term; writes every element)
__global__ void agg_init_kernel(const float* __restrict__ M, const float* __restrict__ dinv,
                                const float* __restrict__ bias, float* __restrict__ agg,
                                int N, int F) {
  size_t i = (size_t)blockIdx.x * blockDim.x + threadIdx.x;
  size_t total = (size_t)N * F;
  if (i >= total) return;
  int node = (int)(i / F);
  int f = (int)(i % F);
  float d = dinv[node];
  agg[i] = d * d * M[i] + bias[f];
}

// one block per edge, blockDim == F: agg[col,f] += norm_e * m[row,f]
__global__ void edge_scatter_kernel(const int* __restrict__ ei, const float* __restrict__ normE,
                                    const float* __restrict__ M, float* __restrict__ agg,
                                    int E, int F) {
  int e = blockIdx.x;
  int f = threadIdx.x;
  int r = ei[e];
  int c = ei[E + e];
  float w = normE[e];
  atomicAdd(&agg[(size_t)c * F + f], w * M[(size_t)r * F + f]);
}

__global__ void reparam_kernel(const float* __restrict__ mu, const float* __restrict__ lv,
                               const float* __restrict__ eps, float* __restrict__ amvo, size_t n) {
  size_t i = (size_t)blockIdx.x * blockDim.x + threadIdx.x;
  if (i < n) amvo[i] = mu[i] + eps[i] * expf(0.5f * lv[i]);
}

// segment max; values are ReLU'd (>=0) so int-bit atomicMax == float max
__global__ void pool_kernel(const float* __restrict__ H, const int* __restrict__ batch,
                            float* __restrict__ X2, int F) {
  int n = blockIdx.x;
  int f = threadIdx.x;
  int b = batch[n];
  float v = H[(size_t)n * F + f];
  atomicMax((int*)&X2[(size_t)b * F + f], __float_as_int(v));
}

// ---------------- weight repack: W[K,Nf] -> WMMA-native packed layout ----------------
// vec4 index (k8, c, hi) -> rows (8*k8 + 2*hi + {0,1}) and (8*k8 + 4 + 2*hi + {0,1}) at col c.
// A lane then fetches its B fragments for TWO k-steps with one global_load_b128.
__global__ void pack_w_kernel(const float* __restrict__ W, float* __restrict__ Wp, int K, int Nf) {
  size_t i = (size_t)blockIdx.x * blockDim.x + threadIdx.x;
  size_t total = (size_t)(K >> 3) * Nf * 2;
  if (i >= total) return;
  int hi = (int)(i & 1);
  size_t t = i >> 1;
  int c  = (int)(t % Nf);
  int k8 = (int)(t / Nf);
  int kb = k8 * 8 + hi * 2;
  v4f v;
  v.x = W[(size_t)(kb + 0) * Nf + c];
  v.y = W[(size_t)(kb + 1) * Nf + c];
  v.z = W[(size_t)(kb + 4) * Nf + c];
  v.w = W[(size_t)(kb + 5) * Nf + c];
  ((v4f*)Wp)[i] = v;
}

// ---------------- f32 WMMA GEMM: Out[M,Nf] = A[M,K] @ W[K,Nf] (+bias)(+relu) --------------
// One wave computes a 16x64 strip via v_wmma_f32_16x16x4_f32, two K-steps per iteration.
// All 8 waves of a block share the same 64-column group -> identical B addresses (L0 reuse).
// Requires M%16==0, K%8==0, Nf%64==0 (all true for every GEMM here).
// EPI: 0 = plain, 1 = +bias +relu, 2 = +bias

template <int EPI>
__global__ __launch_bounds__(256) void gemm_wmma_f32(
    const float* __restrict__ A, const float* __restrict__ Wp,
    const float* __restrict__ bias, float* __restrict__ Out,
    int M, int K, int Nf) {
  int lane = threadIdx.x & 31;
  int wid  = threadIdx.x >> 5;
  int Ng   = Nf >> 6;                          // 64-wide column groups
  int bm   = blockIdx.x / Ng;
  int tn   = blockIdx.x - bm * Ng;
  int tm   = bm * 8 + wid;                     // 8 row tiles per block, shared tn
  if (tm * 16 >= M) return;
  int r0 = tm << 4;
  int c0 = tn << 6;
  int row = lane & 15;                         // M index (A) / N index (B,D)
  int hi  = lane >> 4;                         // half-wave select
  int kb  = hi << 1;                           // K sub-offset {0,2}

  v8f acc0 = {}, acc1 = {}, acc2 = {}, acc3 = {};
  const float* arow  = A + (size_t)(r0 + row) * K + kb;
  const float* bbase = Wp + ((size_t)(c0 + row) * 2 + hi) * 4;   // k8 = 0
  const size_t bstride = (size_t)Nf * 8;                         // floats per k8 step

  for (int k8 = 0; k8 < (K >> 3); ++k8) {
    v2f a0 = *(const v2f*)(arow + (size_t)k8 * 8);
    v2f a1 = *(const v2f*)(arow + (size_t)k8 * 8 + 4);
    const float* bp = bbase + (size_t)k8 * bstride;
    v4f b0 = *(const v4f*)(bp);          // cols c0+row       (two k-steps)
    v4f b1 = *(const v4f*)(bp + 128);    // cols +16  (16*2*4 floats)
    v4f b2 = *(const v4f*)(bp + 256);    // cols +32
    v4f b3 = *(const v4f*)(bp + 384);    // cols +48
    v2f b0l = {b0.x, b0.y}, b0h = {b0.z, b0.w};
    v2f b1l = {b1.x, b1.y}, b1h = {b1.z, b1.w};
    v2f b2l = {b2.x, b2.y}, b2h = {b2.z, b2.w};
    v2f b3l = {b3.x, b3.y}, b3h = {b3.z, b3.w};
    acc0 = __builtin_amdgcn_wmma_f32_16x16x4_f32(false, a0, false, b0l, (short)0, acc0, false, false);
    acc1 = __builtin_amdgcn_wmma_f32_16x16x4_f32(false, a0, false, b1l, (short)0, acc1, false, false);
    acc2 = __builtin_amdgcn_wmma_f32_16x16x4_f32(false, a0, false, b2l, (short)0, acc2, false, false);
    acc3 = __builtin_amdgcn_wmma_f32_16x16x4_f32(false, a0, false, b3l, (short)0, acc3, false, false);
    acc0 = __builtin_amdgcn_wmma_f32_16x16x4_f32(false, a1, false, b0h, (short)0, acc0, false, false);
    acc1 = __builtin_amdgcn_wmma_f32_16x16x4_f32(false, a1, false, b1h, (short)0, acc1, false, false);
    acc2 = __builtin_amdgcn_wmma_f32_16x16x4_f32(false, a1, false, b2h, (short)0, acc2, false, false);
    acc3 = __builtin_amdgcn_wmma_f32_16x16x4_f32(false, a1, false, b3h, (short)0, acc3, false, false);
  }

#pragma unroll
  for (int v = 0; v < 8; ++v) {
    int rr = r0 + v + (hi << 3);               // D VGPR v -> row v (+8 for upper half)
    float* orow = Out + (size_t)rr * Nf + c0 + row;
    float x0 = acc0[v], x1 = acc1[v], x2 = acc2[v], x3 = acc3[v];
    if (EPI >= 1) {
      x0 += bias[c0 + row];
      x1 += bias[c0 + row + 16];
      x2 += bias[c0 + row + 32];
      x3 += bias[c0 + row + 48];
    }
    if (EPI == 1) {
      x0 = fmaxf(x0, 0.0f); x1 = fmaxf(x1, 0.0f);
      x2 = fmaxf(x2, 0.0f); x3 = fmaxf(x3, 0.0f);
    }
    orow[0]  = x0;
    orow[16] = x1;
    orow[32] = x2;
    orow[48] = x3;
  }
}

static inline int cdiv(size_t a, int b) { return (int)((a + b - 1) / b); }

// repack W then run GEMM (stream-ordered, capture-safe)
static inline void launch_gemm(int epi, const float* A, const float* W, float* Wp,
                               const float* bias, float* Out,
                               int M, int K, int Nf, hipStream_t s) {
  size_t packs = (size_t)(K >> 3) * Nf * 2;
  pack_w_kernel<<<cdiv(packs, 256), 256, 0, s>>>(W, Wp, K, Nf);
  int Mt = M / 16;
  int blocks = cdiv((size_t)Mt, 8) * (Nf / 64);
  if (epi == 0)      gemm_wmma_f32<0><<<blocks, 256, 0, s>>>(A, Wp, bias, Out, M, K, Nf);
  else if (epi == 1) gemm_wmma_f32<1><<<blocks, 256, 0, s>>>(A, Wp, bias, Out, M, K, Nf);
  else               gemm_wmma_f32<2><<<blocks, 256, 0, s>>>(A, Wp, bias, Out, M, K, Nf);
}

// ---------------- host orchestration ----------------

extern "C" void kernel_launch(void* const* d_in, const int* in_sizes, int n_in,
                              void* d_out, int out_size, void* d_ws, size_t ws_size,
                              hipStream_t stream) {
  const float* x    = (const float*)d_in[0];
  const int*   ei   = (const int*)  d_in[1];
  const int*   batch= (const int*)  d_in[2];
  const float* eps  = (const float*)d_in[3];
  const float* W1   = (const float*)d_in[4];  const float* b1  = (const float*)d_in[5];
  const float* W2   = (const float*)d_in[6];  const float* b2  = (const float*)d_in[7];
  const float* W3   = (const float*)d_in[8];  const float* b3  = (const float*)d_in[9];
  const float* Wmu  = (const float*)d_in[10]; const float* bmu = (const float*)d_in[11];
  const float* Wlv  = (const float*)d_in[12]; const float* blv = (const float*)d_in[13];
  const float* fc1w = (const float*)d_in[14]; const float* fc1b= (const float*)d_in[15];
  const float* fc2w = (const float*)d_in[16]; const float* fc2b= (const float*)d_in[17];

  const int N   = in_sizes[2];          // 100000
  const int E   = in_sizes[1] / 2;      // 400000
  const int F0  = in_sizes[0] / N;      // 64
  const int F1  = 2 * F0;               // 128
  const int F2  = 3 * F0;               // 192
  const int F4  = 4 * F0;               // 256
  const int HID = in_sizes[15];         // 1024
  const int FIN = in_sizes[17];         // 128
  const int B   = (int)(((size_t)out_size - 3ull * N * F4) / FIN);  // 512

  // workspace layout (floats); all offsets stay 16B aligned
  float* ws    = (float*)d_ws;
  size_t off   = 0;
  float* dinv  = ws + off; off += (size_t)N;
  float* normE = ws + off; off += (size_t)E;
  float* bufA  = ws + off; off += (size_t)N * F4;   // matmul outputs
  float* bufB  = ws + off; off += (size_t)N * F4;   // h1 / h3
  float* bufC  = ws + off; off += (size_t)N * F4;   // h2
  float* x2    = ws + off; off += (size_t)B * F4;
  float* hid   = ws + off; off += (size_t)B * HID;
  float* Wp    = ws + off; off += (size_t)F4 * HID; // packed-weight scratch (max 256x1024)
  (void)ws_size;

  // output layout: amvo [N,F4] | mu [N,F4] | logvar [N,F4] | pmvo [B,FIN]
  float* out  = (float*)d_out;
  float* amvo = out;
  float* mu   = out + (size_t)N * F4;
  float* lv   = out + 2ull * N * F4;
  float* pmvo = out + 3ull * N * F4;

  // --- degree / norm ---
  fill_kernel<<<cdiv(N, 256), 256, 0, stream>>>(dinv, 1.0f, (size_t)N);   // self-loop count
  deg_kernel<<<cdiv(E, 256), 256, 0, stream>>>(ei, dinv, E);
  rsqrt_kernel<<<cdiv(N, 256), 256, 0, stream>>>(dinv, N);
  norm_kernel<<<cdiv(E, 256), 256, 0, stream>>>(ei, dinv, normE, E);

  // --- GCN layer 1: x[N,F0] -> h1[N,F1] in bufB ---
  launch_gemm(0, x, W1, Wp, nullptr, bufA, N, F0, F1, stream);
  agg_init_kernel<<<cdiv((size_t)N * F1, 256), 256, 0, stream>>>(bufA, dinv, b1, bufB, N, F1);
  edge_scatter_kernel<<<E, F1, 0, stream>>>(ei, normE, bufA, bufB, E, F1);
  relu_kernel<<<cdiv((size_t)N * F1, 256), 256, 0, stream>>>(bufB, (size_t)N * F1);

  // --- GCN layer 2: h1 -> h2[N,F2] in bufC ---
  launch_gemm(0, bufB, W2, Wp, nullptr, bufA, N, F1, F2, stream);
  agg_init_kernel<<<cdiv((size_t)N * F2, 256), 256, 0, stream>>>(bufA, dinv, b2, bufC, N, F2);
  edge_scatter_kernel<<<E, F2, 0, stream>>>(ei, normE, bufA, bufC, E, F2);
  relu_kernel<<<cdiv((size_t)N * F2, 256), 256, 0, stream>>>(bufC, (size_t)N * F2);

  // --- GCN layer 3: h2 -> h3[N,F4] in bufB ---
  launch_gemm(0, bufC, W3, Wp, nullptr, bufA, N, F2, F4, stream);
  agg_init_kernel<<<cdiv((size_t)N * F4, 256), 256, 0, stream>>>(bufA, dinv, b3, bufB, N, F4);
  edge_scatter_kernel<<<E, F4, 0, stream>>>(ei, normE, bufA, bufB, E, F4);
  relu_kernel<<<cdiv((size_t)N * F4, 256), 256, 0, stream>>>(bufB, (size_t)N * F4);

  // --- mu head: h3 -> mu (direct to output) ---
  launch_gemm(0, bufB, Wmu, Wp, nullptr, bufA, N, F4, F4, stream);
  agg_init_kernel<<<cdiv((size_t)N * F4, 256), 256, 0, stream>>>(bufA, dinv, bmu, mu, N, F4);
  edge_scatter_kernel<<<E, F4, 0, stream>>>(ei, normE, bufA, mu, E, F4);

  // --- logvar head: h3 -> lv (direct to output) ---
  launch_gemm(0, bufB, Wlv, Wp, nullptr, bufA, N, F4, F4, stream);
  agg_init_kernel<<<cdiv((size_t)N * F4, 256), 256, 0, stream>>>(bufA, dinv, blv, lv, N, F4);
  edge_scatter_kernel<<<E, F4, 0, stream>>>(ei, normE, bufA, lv, E, F4);

  // --- reparameterize ---
  reparam_kernel<<<cdiv((size_t)N * F4, 256), 256, 0, stream>>>(mu, lv, eps, amvo, (size_t)N * F4);

  // --- global max pool (h3 in bufB, ReLU'd so >= 0) ---
  fill_kernel<<<cdiv((size_t)B * F4, 256), 256, 0, stream>>>(x2, 0.0f, (size_t)B * F4);
  pool_kernel<<<N, F4, 0, stream>>>(bufB, batch, x2, F4);

  // --- MLP head: relu(x2 @ fc1 + b) @ fc2 + b ---
  launch_gemm(1, x2, fc1w, Wp, fc1b, hid, B, F4, HID, stream);
  launch_gemm(2, hid, fc2w, Wp, fc2b, pmvo, B, HID, FIN, stream);
}